// Attention3_58866821759172
// MI455X (gfx1250) — compile-verified
//
#include <hip/hip_runtime.h>
#include <math.h>

// ---------------------------------------------------------------------------
// Attention3: h -> (4x GEMM) -> att_h; tanh(p_att + att_h)·Wa -> scores;
// masked softmax over S; att_res = weight · att_feats.
// B=128, S=1024, RNN=2048, HID=512. Memory-bound (~1.3 GB streamed).
// ---------------------------------------------------------------------------

typedef __attribute__((ext_vector_type(2))) float v2f;
typedef __attribute__((ext_vector_type(8))) float v8f;

#define ATT_B   128
#define ATT_S   1024
#define ATT_RNN 2048
#define ATT_HID 512
#define MIN_VALUE -100000000.0f

// ---------------------------------------------------------------------------
// Kernel 1: Y[M,N] = X[M,K] @ W[N,K]^T + bias[N]   via V_WMMA_F32_16X16X4_F32
// One wave (32 threads) per 16x16 output tile. Wave32, EXEC all-ones.
//
// f32 16x4 A layout (ISA 7.12.2): lanes 0-15 -> M=lane, VGPR0=K0 VGPR1=K1;
//                                 lanes 16-31 -> M=lane-16, VGPR0=K2 VGPR1=K3.
// B (4x16) mirrors with lane = N. Since B_wmma[k][n] = W[n][k] and W is
// row-major [N,K], both fragments are two consecutive-K loads from a row.
// D layout: VGPR r -> M = m0 + r (+8 for hi lanes), N = n0 + (lane & 15).
// ---------------------------------------------------------------------------
__global__ void __launch_bounds__(32)
gemm_wmma_f32(const float* __restrict__ X, const float* __restrict__ W,
              const float* __restrict__ bias, float* __restrict__ Y,
              int M, int N, int K) {
    const int lane = threadIdx.x;            // 0..31 (wave32)
    const int m0   = blockIdx.y * 16;
    const int n0   = blockIdx.x * 16;
    const bool hi  = lane >= 16;
    const int r16  = lane & 15;
    const int koff = hi ? 2 : 0;

    const float* __restrict__ Arow = X + (size_t)(m0 + r16) * K;
    const float* __restrict__ Brow = W + (size_t)(n0 + r16) * K;

    v8f acc = {};
    for (int k = 0; k < K; k += 4) {
        v2f a = { Arow[k + koff], Arow[k + koff + 1] };
        v2f b = { Brow[k + koff], Brow[k + koff + 1] };
        // 8 args: (neg_a, A, neg_b, B, c_mod, C, reuse_a, reuse_b)
        acc = __builtin_amdgcn_wmma_f32_16x16x4_f32(
            false, a, false, b, (short)0, acc, false, false);
    }

    const float bn    = bias[n0 + r16];
    const int   mbase = m0 + (hi ? 8 : 0);
#pragma unroll
    for (int r = 0; r < 8; ++r) {
        Y[(size_t)(mbase + r) * N + (n0 + r16)] = acc[r] + bn;
    }
}

// ---------------------------------------------------------------------------
// Kernel 2: scores[b,s] = sum_h tanh(p_att[b,s,h] + att_h[b,h]) * Wa[h] + ba
// One wave32 per (b,s); lanes stride HID, xor-shuffle reduction.
// ---------------------------------------------------------------------------
__global__ void __launch_bounds__(256)
scores_kernel(const float* __restrict__ p_att, const float* __restrict__ att_h,
              const float* __restrict__ Wa, const float* __restrict__ ba,
              float* __restrict__ scores) {
    const int wid  = blockIdx.x * (blockDim.x >> 5) + (threadIdx.x >> 5);
    const int lane = threadIdx.x & 31;
    const int b = wid / ATT_S;
    const int s = wid % ATT_S;

    const float* __restrict__ p  = p_att + ((size_t)b * ATT_S + s) * ATT_HID;
    const float* __restrict__ ah = att_h + (size_t)b * ATT_HID;

    float partial = 0.0f;
#pragma unroll 4
    for (int h = lane; h < ATT_HID; h += 32) {
        partial += tanhf(p[h] + ah[h]) * Wa[h];
    }
#pragma unroll
    for (int off = 16; off > 0; off >>= 1) {
        partial += __shfl_xor(partial, off, 32);
    }
    if (lane == 0) {
        scores[(size_t)b * ATT_S + s] = partial + ba[0];
    }
}

// ---------------------------------------------------------------------------
// Kernel 3: masked softmax over S for one batch per block.
// mask != 0 -> MIN_VALUE (reference: where(mask, MIN_VALUE, scores)).
// ---------------------------------------------------------------------------
__global__ void __launch_bounds__(256)
softmax_kernel(const float* __restrict__ scores, const int* __restrict__ mask,
               float* __restrict__ weight) {
    __shared__ float red[256];
    const int b = blockIdx.x;
    const int t = threadIdx.x;
    const int PER = ATT_S / 256;   // 4

    float local[PER];
    float mx = -INFINITY;
#pragma unroll
    for (int i = 0; i < PER; ++i) {
        const int s = t + i * 256;
        float v = scores[(size_t)b * ATT_S + s];
        if (mask[(size_t)b * ATT_S + s] != 0) v = MIN_VALUE;
        local[i] = v;
        mx = fmaxf(mx, v);
    }
    red[t] = mx;
    __syncthreads();
    for (int st = 128; st > 0; st >>= 1) {
        if (t < st) red[t] = fmaxf(red[t], red[t + st]);
        __syncthreads();
    }
    mx = red[0];
    __syncthreads();

    float sum = 0.0f;
#pragma unroll
    for (int i = 0; i < PER; ++i) {
        local[i] = __expf(local[i] - mx);
        sum += local[i];
    }
    red[t] = sum;
    __syncthreads();
    for (int st = 128; st > 0; st >>= 1) {
        if (t < st) red[t] += red[t + st];
        __syncthreads();
    }
    const float inv = 1.0f / red[0];

#pragma unroll
    for (int i = 0; i < PER; ++i) {
        weight[(size_t)b * ATT_S + (t + i * 256)] = local[i] * inv;
    }
}

// ---------------------------------------------------------------------------
// Kernel 4: att_res[b,r] = sum_s weight[b,s] * att_feats[b,s,r]
// The 1 GB stream: blockIdx.y = batch, blockIdx.x = 1024-wide R chunk,
// float4 per thread, fully coalesced along R; prefetch the next S rows.
// ---------------------------------------------------------------------------
__global__ void __launch_bounds__(256)
attres_kernel(const float* __restrict__ weight, const float* __restrict__ att_feats,
              float* __restrict__ out) {
    const int b  = blockIdx.y;
    const int r0 = blockIdx.x * 1024 + threadIdx.x * 4;

    const float* __restrict__ wrow = weight + (size_t)b * ATT_S;
    const float* __restrict__ base =
        att_feats + (size_t)b * ATT_S * ATT_RNN + r0;

    float4 acc = make_float4(0.f, 0.f, 0.f, 0.f);
    for (int s = 0; s < ATT_S; ++s) {
        __builtin_prefetch(base + (size_t)(s + 8) * ATT_RNN, 0, 0);
        const float  w = wrow[s];
        const float4 f = *(const float4*)(base + (size_t)s * ATT_RNN);
        acc.x += w * f.x;
        acc.y += w * f.y;
        acc.z += w * f.z;
        acc.w += w * f.w;
    }
    *(float4*)(out + (size_t)b * ATT_RNN + r0) = acc;
}

// ---------------------------------------------------------------------------
extern "C" void kernel_launch(void* const* d_in, const int* in_sizes, int n_in,
                              void* d_out, int out_size, void* d_ws, size_t ws_size,
                              hipStream_t stream) {
    const float* h         = (const float*)d_in[0];
    const float* att_feats = (const float*)d_in[1];
    const float* p_att     = (const float*)d_in[2];
    const int*   mask      = (const int*)  d_in[3];
    const float* W1 = (const float*)d_in[4];
    const float* b1 = (const float*)d_in[5];
    const float* W2 = (const float*)d_in[6];
    const float* b2 = (const float*)d_in[7];
    const float* W3 = (const float*)d_in[8];
    const float* b3 = (const float*)d_in[9];
    const float* W4 = (const float*)d_in[10];
    const float* b4 = (const float*)d_in[11];
    const float* Wa = (const float*)d_in[12];
    const float* ba = (const float*)d_in[13];
    float* out = (float*)d_out;

    // Workspace layout (floats): t1[128,1024] t2[128,1024] t3[128,512]
    //                            att_h[128,512] scores[128,1024] weight[128,1024]
    float* t1     = (float*)d_ws;
    float* t2     = t1     + (size_t)ATT_B * 1024;
    float* t3     = t2     + (size_t)ATT_B * 1024;
    float* atth   = t3     + (size_t)ATT_B * 512;
    float* scores = atth   + (size_t)ATT_B * 512;
    float* weight = scores + (size_t)ATT_B * ATT_S;

    // --- Stage 1: att_h = (((h W1^T+b1) W2^T+b2) W3^T+b3) W4^T+b4 ---
    {
        dim3 blk(32, 1, 1);
        gemm_wmma_f32<<<dim3(1024 / 16, ATT_B / 16), blk, 0, stream>>>(
            h,  W1, b1, t1,   ATT_B, 1024, ATT_RNN);
        gemm_wmma_f32<<<dim3(1024 / 16, ATT_B / 16), blk, 0, stream>>>(
            t1, W2, b2, t2,   ATT_B, 1024, 1024);
        gemm_wmma_f32<<<dim3(512  / 16, ATT_B / 16), blk, 0, stream>>>(
            t2, W3, b3, t3,   ATT_B, 512, 1024);
        gemm_wmma_f32<<<dim3(512  / 16, ATT_B / 16), blk, 0, stream>>>(
            t3, W4, b4, atth, ATT_B, 512, 512);
    }

    // --- Stage 2: scores = tanh(p_att + att_h) . Wa + ba ---
    {
        const int waves  = ATT_B * ATT_S;          // one wave32 per (b,s)
        const int blocks = waves / 8;              // 8 waves per 256-thread block
        scores_kernel<<<blocks, 256, 0, stream>>>(p_att, atth, Wa, ba, scores);
    }

    // --- Stage 3: masked softmax over S ---
    softmax_kernel<<<ATT_B, 256, 0, stream>>>(scores, mask, weight);

    // --- Stage 4: att_res = weight . att_feats  (1 GB stream) ---
    attres_kernel<<<dim3(ATT_RNN / 1024, ATT_B), 256, 0, stream>>>(
        weight, att_feats, out);
}